// BoundaryDistillationLoss_87230785781774
// MI455X (gfx1250) — compile-verified
//
#include <hip/hip_runtime.h>
#include <hip/hip_bf16.h>

// Problem dims (fixed by reference)
#define B_  4
#define C_  19
#define H_  512
#define W_  1024

// Tiling: 32x8 outputs per block, halo = 40 cols (aligned) x 10 rows
#define TW   32
#define TH   8
#define HWW  40                   // halo width, 16B-chunk aligned (w0-4 .. w0+35)
#define ROWS (TH + 2)             // 10 halo rows
#define RSTRIDE HWW               // LDS row stride in words (160 B, 16B aligned)
#define PLANE (ROWS * RSTRIDE)    // 400 words per (tensor,channel) plane
#define WORDS (2 * C_ * PLANE)    // 15200 fp32 words in LDS
#define CHUNKS (2 * C_ * ROWS * (HWW / 4))  // 3800 16-byte chunks
#define SCOLS 34                  // columns actually needed by stencil (lc 3..36)
#define NTHREADS 256

typedef __attribute__((address_space(3))) void lds_void_t;

__device__ __forceinline__ unsigned lds_addr_of(const void* p) {
    return (unsigned)(unsigned long long)(lds_void_t*)(void*)p;
}

__global__ void zero_out_kernel(float* out) {
    if (threadIdx.x == 0 && blockIdx.x == 0) out[0] = 0.0f;
}

__global__ void __launch_bounds__(NTHREADS)
boundary_distill_kernel(const float* __restrict__ S,
                        const float* __restrict__ T,
                        float* __restrict__ out) {
    __shared__ float tile[WORDS];   // [t][c][hh][lc]; logits, then probs in place
    __shared__ float red[NTHREADS];

    const int tid = threadIdx.x;
    const int w0  = blockIdx.x * TW;
    const int h0  = blockIdx.y * TH;
    const int b   = blockIdx.z;

    // ---- Phase 1: async global -> LDS, 16B chunks (B128), zero OOB chunks --
    for (int k = tid; k < CHUNKS; k += NTHREADS) {
        int cw = k % (HWW / 4);        // chunk within row
        int r  = k / (HWW / 4);
        int hh = r % ROWS;
        int r2 = r / ROWS;
        int c  = r2 % C_;
        int t  = r2 / C_;
        int h  = h0 + hh - 1;
        int ws = w0 - 4 + cw * 4;      // global start col of this chunk
        int idx = (t * C_ + c) * PLANE + hh * RSTRIDE + cw * 4;
        bool inr = (h >= 0) && (h < H_) && (ws >= 0) && (ws < W_);
        if (inr) {
            const float* gp = (t ? T : S) +
                ((((size_t)b * C_ + c) * H_ + h) * (size_t)W_ + ws);
            unsigned la = lds_addr_of(&tile[idx]);
            // CDNA5 async copy: copies 16 bytes from the global address gp to the LDS address la; tracked by ASYNCcnt.
            asm volatile("global_load_async_to_lds_b128 %0, %1, off"
                         :: "v"(la), "v"(gp) : "memory");
        } else {
            *(float4*)&tile[idx] = make_float4(0.f, 0.f, 0.f, 0.f); // zero pad
        }
    }
    asm volatile("s_wait_asynccnt 0x0" ::: "memory");
    __syncthreads();

    // ---- Phase 2: in-place softmax over C (only the 34 stencil columns) ----
    for (int e = tid; e < 2 * ROWS * SCOLS; e += NTHREADS) {
        int t  = e / (ROWS * SCOLS);
        int p  = e % (ROWS * SCOLS);
        int hh = p / SCOLS;
        int lc = p % SCOLS + 3;        // local col 3..36
        int h  = h0 + hh - 1;
        int w  = w0 + (lc - 4);
        if (h < 0 || h >= H_ || w < 0 || w >= W_) continue;  // stays zero
        int base = t * C_ * PLANE + hh * RSTRIDE + lc;
        float v[C_];
        float m = -3.4e38f;
        #pragma unroll
        for (int c = 0; c < C_; ++c) {
            v[c] = tile[base + c * PLANE];
            m = fmaxf(m, v[c]);
        }
        float s = 0.0f;
        #pragma unroll
        for (int c = 0; c < C_; ++c) {
            v[c] = __expf(v[c] - m);
            s += v[c];
        }
        float inv = 1.0f / s;
        #pragma unroll
        for (int c = 0; c < C_; ++c) tile[base + c * PLANE] = v[c] * inv;
    }
    __syncthreads();

    // ---- Phase 3: Sobel via vertical sliding window (6 LDS reads / diff) ---
    float acc = 0.0f;
    for (int u = tid; u < C_ * TW; u += NTHREADS) {
        int ow = u % TW;               // consecutive lanes -> consecutive cols
        int c  = u / TW;
        int lc0 = ow + 3;              // left column of 3x3 window
        const float* qs = &tile[c * PLANE + lc0];
        const float* qt = &tile[(C_ + c) * PLANE + lc0];

        // preload rows 0 and 1 of the halo for both tensors
        float sa0 = qs[0], sa1 = qs[1], sa2 = qs[2];
        float sb0 = qs[RSTRIDE], sb1 = qs[RSTRIDE + 1], sb2 = qs[RSTRIDE + 2];
        float ta0 = qt[0], ta1 = qt[1], ta2 = qt[2];
        float tb0 = qt[RSTRIDE], tb1 = qt[RSTRIDE + 1], tb2 = qt[RSTRIDE + 2];

        #pragma unroll
        for (int oh = 0; oh < TH; ++oh) {
            int ro = (oh + 2) * RSTRIDE;
            float sc0 = qs[ro], sc1 = qs[ro + 1], sc2 = qs[ro + 2];
            float tc0 = qt[ro], tc1 = qt[ro + 1], tc2 = qt[ro + 2];

            float gxs = (sa2 - sa0) + 2.f * (sb2 - sb0) + (sc2 - sc0);
            float gys = (sc0 + 2.f * sc1 + sc2) - (sa0 + 2.f * sa1 + sa2);
            float gxt = (ta2 - ta0) + 2.f * (tb2 - tb0) + (tc2 - tc0);
            float gyt = (tc0 + 2.f * tc1 + tc2) - (ta0 + 2.f * ta1 + ta2);

            float d = sqrtf(gxs * gxs + gys * gys) - sqrtf(gxt * gxt + gyt * gyt);
            acc += d * d;

            sa0 = sb0; sa1 = sb1; sa2 = sb2;  sb0 = sc0; sb1 = sc1; sb2 = sc2;
            ta0 = tb0; ta1 = tb1; ta2 = tb2;  tb0 = tc0; tb1 = tc1; tb2 = tc2;
        }
    }

    // ---- Block reduction + one atomic per block ----------------------------
    red[tid] = acc;
    __syncthreads();
    for (int s = NTHREADS / 2; s > 0; s >>= 1) {
        if (tid < s) red[tid] += red[tid + s];
        __syncthreads();
    }
    if (tid == 0) {
        const float invN = 1.0f / (float)((long long)B_ * C_ * H_ * W_);
        atomicAdd(out, red[0] * invN);
    }
}

extern "C" void kernel_launch(void* const* d_in, const int* in_sizes, int n_in,
                              void* d_out, int out_size, void* d_ws, size_t ws_size,
                              hipStream_t stream) {
    const float* S = (const float*)d_in[0];
    const float* T = (const float*)d_in[1];
    float* out = (float*)d_out;

    zero_out_kernel<<<1, 1, 0, stream>>>(out);

    dim3 grid(W_ / TW, H_ / TH, B_);  // 32 x 64 x 4 = 8192 blocks
    boundary_distill_kernel<<<grid, NTHREADS, 0, stream>>>(S, T, out);
}